// TripletAttention_57810259804441
// MI455X (gfx1250) — compile-verified
//
#include <hip/hip_runtime.h>
#include <stdint.h>

// TripletAttention, fused two-pass, fp32, HBM-roofline design (~768 MB total traffic).
// K1 (per (b,c) 128x128 plane in LDS): pool over H -> a_h(b,c,w), pool over W -> a_w(b,c,h),
//    7-tap conv + BN + sigmoid fused in-block. 256 threads; reductions split across 2 halves.
// K2 (per (b,h) 256x128 slab in LDS, needs CDNA5's 320KB LDS): pool over C -> a_c(b,h,w)
//    fused, then out = x * (a_h + a_w + a_c)/3 from the resident slab. x read exactly twice.

#define BDIM 16
#define CDIM 256
#define HDIM 128
#define WDIM 128
#define BN_EPS 1e-5f
#define NEG_BIG (-3.402823466e38f)

// ---- CDNA5 async global->LDS copy (ASYNCcnt path) ----
__device__ __forceinline__ void async_load16(const float* gsrc, unsigned lds_byte_off) {
  // global_load_async_to_lds_b128 vDst(LDS byte offset), vAddr(64b global), off
  asm volatile("global_load_async_to_lds_b128 %0, %1, off"
               :: "v"(lds_byte_off), "v"(gsrc)
               : "memory");
}
__device__ __forceinline__ void wait_async0() {
  asm volatile("s_wait_asynccnt 0" ::: "memory");
}
__device__ __forceinline__ unsigned lds_off(const void* p) {
  // low 32 bits of a flat shared address == LDS byte offset (ISA 10.2 aperture mapping)
  return (unsigned)(uintptr_t)p;
}
__device__ __forceinline__ float sigmoidf_(float x) {
  return 1.0f / (1.0f + __expf(-x));
}

// ============ Kernel 1: one block per (c,b); produces (1/3)*a_h(b,c,w) and (1/3)*a_w(b,c,h) ============
__global__ __launch_bounds__(256) void ta_reduce_hw(
    const float* __restrict__ x,
    const float* __restrict__ w_h, const float* __restrict__ g_h,
    const float* __restrict__ b_h, const float* __restrict__ m_h,
    const float* __restrict__ v_h,
    const float* __restrict__ w_w, const float* __restrict__ g_w,
    const float* __restrict__ b_w, const float* __restrict__ m_w,
    const float* __restrict__ v_w,
    float* __restrict__ a_h, float* __restrict__ a_w)
{
  extern __shared__ float smem[];
  constexpr int LSTRIDE = WDIM + 4; // 132: rows 16B-aligned for B128, <=2-way LDS bank conflicts
  float* plane = smem;                    // HDIM * LSTRIDE
  float* pS    = smem + HDIM * LSTRIDE;   // 256 (per-half partial sums)
  float* pM    = pS + 2 * WDIM;           // 256 (per-half partial maxes)
  float* sAvg  = pM + 2 * WDIM;           // 128
  float* sMax  = sAvg + WDIM;             // 128

  const int c = blockIdx.x;
  const int b = blockIdx.y;
  const int t = threadIdx.x;
  const int lane128 = t & 127;
  const int half    = t >> 7;       // 0 or 1

  const float* xp = x + ((size_t)b * CDIM + c) * HDIM * WDIM;

  // async-copy the 128x128 plane into padded LDS rows (16 x B128 per lane, 8 waves)
  const int crow = t >> 5;          // 0..7
  const int w4   = (t & 31) * 4;    // 0..124
  #pragma unroll 4
  for (int i = 0; i < HDIM / 8; ++i) {
    const int h = i * 8 + crow;
    async_load16(xp + (size_t)h * WDIM + w4, lds_off(&plane[h * LSTRIDE + w4]));
  }
  wait_async0();
  __syncthreads();

  // ---- pool over H (thread owns column w, two halves of the h-range) ----
  {
    const int w = lane128;
    float s = 0.0f, mx = NEG_BIG;
    #pragma unroll 8
    for (int h = half * (HDIM / 2); h < (half + 1) * (HDIM / 2); ++h) {
      const float v = plane[h * LSTRIDE + w];
      s += v;
      mx = fmaxf(mx, v);
    }
    pS[half * WDIM + w] = s;
    pM[half * WDIM + w] = mx;
  }
  __syncthreads();
  if (t < WDIM) {
    const int w = t;
    sAvg[w] = (pS[w] + pS[WDIM + w]) * (1.0f / HDIM);
    sMax[w] = fmaxf(pM[w], pM[WDIM + w]);
  }
  __syncthreads();
  // conv(1x7 along w, zero pad) + BN + sigmoid -> (1/3)*a_h(b,c,w)
  if (t < WDIM) {
    const int w = t;
    const float scale = g_h[0] * rsqrtf(v_h[0] + BN_EPS);
    const float shift = b_h[0] - m_h[0] * scale;
    float y = 0.0f;
    #pragma unroll
    for (int j = 0; j < 7; ++j) {
      const int q = w + j - 3;
      if (0 <= q && q < WDIM)
        y += w_h[j] * sAvg[q] + w_h[7 + j] * sMax[q];
    }
    a_h[((size_t)b * CDIM + c) * WDIM + w] = sigmoidf_(y * scale + shift) * (1.0f / 3.0f);
  }

  // ---- pool over W (thread owns row h, two halves of the w-range) ----
  {
    const int h = lane128;
    float s = 0.0f, mx = NEG_BIG;
    #pragma unroll 8
    for (int w = half * (WDIM / 2); w < (half + 1) * (WDIM / 2); ++w) {
      const float v = plane[h * LSTRIDE + w];
      s += v;
      mx = fmaxf(mx, v);
    }
    pS[half * HDIM + h] = s;
    pM[half * HDIM + h] = mx;
  }
  __syncthreads();  // also orders conv-A reads of sAvg/sMax before overwrite below
  if (t < HDIM) {
    const int h = t;
    sAvg[h] = (pS[h] + pS[HDIM + h]) * (1.0f / WDIM);
    sMax[h] = fmaxf(pM[h], pM[HDIM + h]);
  }
  __syncthreads();
  // conv(1x7 along h) + BN + sigmoid -> (1/3)*a_w(b,c,h)
  if (t < HDIM) {
    const int h = t;
    const float scale = g_w[0] * rsqrtf(v_w[0] + BN_EPS);
    const float shift = b_w[0] - m_w[0] * scale;
    float y = 0.0f;
    #pragma unroll
    for (int j = 0; j < 7; ++j) {
      const int q = h + j - 3;
      if (0 <= q && q < HDIM)
        y += w_w[j] * sAvg[q] + w_w[7 + j] * sMax[q];
    }
    a_w[((size_t)b * CDIM + c) * HDIM + h] = sigmoidf_(y * scale + shift) * (1.0f / 3.0f);
  }
}

// ============ Kernel 2: one block per (h,b); a_c fused + final multiply from LDS slab ============
__global__ __launch_bounds__(256) void ta_finalize(
    const float* __restrict__ x,
    const float* __restrict__ w_c, const float* __restrict__ g_c,
    const float* __restrict__ b_c, const float* __restrict__ m_c,
    const float* __restrict__ v_c,
    const float* __restrict__ a_h, const float* __restrict__ a_w,
    float* __restrict__ out)
{
  extern __shared__ float smem[];
  float* slab  = smem;                 // CDIM * WDIM = 32768 floats (128 KB; needs CDNA5 LDS)
  float* pS    = smem + CDIM * WDIM;   // 256
  float* pM    = pS + 2 * WDIM;        // 256
  float* scAvg = pM + 2 * WDIM;        // 128
  float* scMax = scAvg + WDIM;         // 128
  float* sAc   = scMax + WDIM;         // 128
  float* sAw   = sAc + WDIM;           // 256

  const int h = blockIdx.x;
  const int b = blockIdx.y;
  const int t = threadIdx.x;
  const int lane128 = t & 127;
  const int half    = t >> 7;

  // async-copy x[b, :, h, :] (CDIM rows of 512B, 64KB apart) into LDS; 32 x B128 per lane
  const int crow = t >> 5;          // 0..7
  const int w4   = (t & 31) * 4;
  const float* xb = x + ((size_t)b * CDIM * HDIM + h) * WDIM;
  #pragma unroll 4
  for (int i = 0; i < CDIM / 8; ++i) {
    const int c = i * 8 + crow;
    async_load16(xb + (size_t)c * HDIM * WDIM + w4, lds_off(&slab[c * WDIM + w4]));
  }
  // overlap the async copy: gather the pre-scaled a_w column for this (b,h)
  sAw[t] = a_w[((size_t)b * CDIM + t) * HDIM + h];  // already holds a_w/3

  wait_async0();
  __syncthreads();

  // ---- pool over C (thread owns column w, two halves of the c-range) ----
  {
    const int w = lane128;
    float s = 0.0f, mx = NEG_BIG;
    #pragma unroll 8
    for (int c = half * (CDIM / 2); c < (half + 1) * (CDIM / 2); ++c) {
      const float v = slab[c * WDIM + w];
      s += v;
      mx = fmaxf(mx, v);
    }
    pS[half * WDIM + w] = s;
    pM[half * WDIM + w] = mx;
  }
  __syncthreads();
  if (t < WDIM) {
    const int w = t;
    scAvg[w] = (pS[w] + pS[WDIM + w]) * (1.0f / CDIM);
    scMax[w] = fmaxf(pM[w], pM[WDIM + w]);
  }
  __syncthreads();
  // conv(1x7 along w) + BN + sigmoid -> (1/3)*a_c(b,h,w)
  if (t < WDIM) {
    const int w = t;
    const float scale = g_c[0] * rsqrtf(v_c[0] + BN_EPS);
    const float shift = b_c[0] - m_c[0] * scale;
    float y = 0.0f;
    #pragma unroll
    for (int j = 0; j < 7; ++j) {
      const int q = w + j - 3;
      if (0 <= q && q < WDIM)
        y += w_c[j] * scAvg[q] + w_c[7 + j] * scMax[q];
    }
    sAc[w] = sigmoidf_(y * scale + shift) * (1.0f / 3.0f);
  }
  __syncthreads();

  // ---- out(b,c,h,w) = x * (a_h/3 + a_w/3 + a_c/3), each half covers 128 c's ----
  {
    const int w  = lane128;
    const float ac = sAc[w];
    const int c0 = half * (CDIM / 2);
    const float* ahp  = a_h + (size_t)b * CDIM * WDIM + w;   // holds a_h/3
    float*       outp = out + ((size_t)b * CDIM * HDIM + h) * WDIM + w;
    #pragma unroll 4
    for (int c = c0; c < c0 + CDIM / 2; ++c) {
      const float v   = slab[c * WDIM + w];
      const float att = ahp[(size_t)c * WDIM] + sAw[c] + ac;
      outp[(size_t)c * HDIM * WDIM] = v * att;
    }
  }
}

extern "C" void kernel_launch(void* const* d_in, const int* in_sizes, int n_in,
                              void* d_out, int out_size, void* d_ws, size_t ws_size,
                              hipStream_t stream) {
  const float* x   = (const float*)d_in[0];
  const float* w_h = (const float*)d_in[1];
  const float* g_h = (const float*)d_in[2];
  const float* b_h = (const float*)d_in[3];
  const float* m_h = (const float*)d_in[4];
  const float* v_h = (const float*)d_in[5];
  const float* w_w = (const float*)d_in[6];
  const float* g_w = (const float*)d_in[7];
  const float* b_w = (const float*)d_in[8];
  const float* m_w = (const float*)d_in[9];
  const float* v_w = (const float*)d_in[10];
  const float* w_c = (const float*)d_in[11];
  const float* g_c = (const float*)d_in[12];
  const float* b_c = (const float*)d_in[13];
  const float* m_c = (const float*)d_in[14];
  const float* v_c = (const float*)d_in[15];
  float* out = (float*)d_out;

  float* a_h = (float*)d_ws;                            // B*C*W floats (pre-scaled by 1/3)
  float* a_w = a_h + (size_t)BDIM * CDIM * WDIM;        // B*C*H floats (pre-scaled by 1/3)

  const size_t lds1 = (size_t)(HDIM * (WDIM + 4) + 6 * WDIM) * sizeof(float);   // ~70.7 KB
  dim3 grid1(CDIM, BDIM);
  ta_reduce_hw<<<grid1, 256, lds1, stream>>>(x, w_h, g_h, b_h, m_h, v_h,
                                             w_w, g_w, b_w, m_w, v_w, a_h, a_w);

  const size_t lds2 = (size_t)(CDIM * WDIM + 4 * WDIM + 4 * WDIM + CDIM) * sizeof(float); // ~136 KB
  dim3 grid2(HDIM, BDIM);
  ta_finalize<<<grid2, 256, lds2, stream>>>(x, w_c, g_c, b_c, m_c, v_c,
                                            a_h, a_w, out);
}